// pairDistLayer_87351044866352
// MI455X (gfx1250) — compile-verified
//
#include <hip/hip_runtime.h>
#include <hip/hip_bf16.h>
#include <math.h>

typedef float v2f __attribute__((ext_vector_type(2)));
typedef float v8f __attribute__((ext_vector_type(8)));

#define NPTS 4096
#define DIM  512
#define BT   128          // output tile is BT x BT
#define KC   32           // k-chunk staged through LDS
#define NCHUNK (DIM / KC)
#define LSTR 36           // LDS row stride in floats (144B: 16B aligned, bank-conflict free)

// ---------------- Kernel 1: squared row norms ----------------
__global__ __launch_bounds__(256) void sqnorm_kernel(const float* __restrict__ x,
                                                     float* __restrict__ sq) {
    int wave = threadIdx.x >> 5;
    int lane = threadIdx.x & 31;
    int row  = blockIdx.x * 8 + wave;
    const float* xr = x + (size_t)row * DIM;
    float s = 0.f;
#pragma unroll
    for (int j = 0; j < DIM / 32; ++j) {
        float v = xr[lane + j * 32];
        s += v * v;
    }
#pragma unroll
    for (int off = 16; off > 0; off >>= 1)
        s += __shfl_xor(s, off, 32);
    if (lane == 0) sq[row] = s;
}

// Async global -> LDS copy of one 16-byte packet (CDNA5, ASYNCcnt-tracked).
__device__ __forceinline__ void async_copy_b128(unsigned lds_byte_addr,
                                                const float* gaddr) {
    asm volatile("global_load_async_to_lds_b128 %0, %1, off"
                 :: "v"(lds_byte_addr), "v"(gaddr)
                 : "memory");
}

__device__ __forceinline__ void wait_async_zero() {
#if __has_builtin(__builtin_amdgcn_s_wait_asynccnt)
    __builtin_amdgcn_s_wait_asynccnt(0);
#else
    asm volatile("s_wait_asynccnt 0x0" ::: "memory");
#endif
}

// ---------------- Kernel 2: tiled distance via fp32 WMMA ----------------
// Grid (N/BT)^2; blocks with bi < bj exit uniformly (EXEC all-ones for WMMA).
// 8 waves; wave (wr, wc) computes a 64x32 subtile = 4x2 accumulators of 16x16.
// Double-buffered LDS, filled by async global->LDS DMA overlapped with WMMA.
__global__ __launch_bounds__(256) void pairdist_kernel(const float* __restrict__ x,
                                                       const float* __restrict__ sq,
                                                       float* __restrict__ out) {
    const int bi = blockIdx.y;
    const int bj = blockIdx.x;
    if (bi < bj) return;
    const int rowBase = bi * BT;
    const int colBase = bj * BT;

    __shared__ __align__(16) float ldsA[2][BT * LSTR];
    __shared__ __align__(16) float ldsB[2][BT * LSTR];

    const int tid  = threadIdx.x;
    const int wave = tid >> 5;
    const int lane = tid & 31;
    const int wr   = wave >> 2;   // 0..1 : 64 rows each
    const int wc   = wave & 3;    // 0..3 : 32 cols each
    const int l15  = lane & 15;
    const int lhi  = lane >> 4;   // 0/1 : K pair select (A/B frags), M+8 (C/D)

    // LDS byte addresses (low 32 bits of the generic pointer = LDS offset).
    const unsigned ldsA_base[2] = {(unsigned)(size_t)&ldsA[0][0],
                                   (unsigned)(size_t)&ldsA[1][0]};
    const unsigned ldsB_base[2] = {(unsigned)(size_t)&ldsB[0][0],
                                   (unsigned)(size_t)&ldsB[1][0]};

    // Per-thread staging slots: 128 rows x 8 float4 = 1024 slots; 4/thread/tile.
    auto stage = [&](int kc, int buf) {
#pragma unroll
        for (int s = 0; s < 4; ++s) {
            int slot = tid + s * 256;   // 0..1023
            int r = slot >> 3;          // row within tile
            int q = slot & 7;           // float4 index within 32-float chunk
            const float* ga = x + (size_t)(rowBase + r) * DIM + kc + q * 4;
            const float* gb = x + (size_t)(colBase + r) * DIM + kc + q * 4;
            unsigned la = ldsA_base[buf] + (unsigned)(r * LSTR + q * 4) * 4u;
            unsigned lb = ldsB_base[buf] + (unsigned)(r * LSTR + q * 4) * 4u;
            async_copy_b128(la, ga);
            async_copy_b128(lb, gb);
        }
    };

    v8f acc[4][2];
    const v8f vzero = {0.f, 0.f, 0.f, 0.f, 0.f, 0.f, 0.f, 0.f};
#pragma unroll
    for (int mt = 0; mt < 4; ++mt)
#pragma unroll
        for (int nt = 0; nt < 2; ++nt)
            acc[mt][nt] = vzero;

    stage(0, 0);   // prefetch chunk 0

    for (int c = 0; c < NCHUNK; ++c) {
        const int cur = c & 1;
        // Chunk c's async loads complete (own wave), then sync all waves.
        wait_async_zero();
        __syncthreads();
        // Kick off DMA for the next chunk into the other buffer; it overlaps
        // with the WMMA work below. The buffer it targets was last *read*
        // in iteration c-1, which finished before the barrier above.
        if (c + 1 < NCHUNK) stage((c + 1) * KC, cur ^ 1);

        const float* la = ldsA[cur];
        const float* lb = ldsB[cur];
#pragma unroll
        for (int kk = 0; kk < KC; kk += 4) {
            const int kOff = kk + lhi * 2;
            v2f afrag[4], bfrag[2];
#pragma unroll
            for (int mt = 0; mt < 4; ++mt) {
                int r = wr * 64 + mt * 16 + l15;
                afrag[mt] = *(const v2f*)(&la[r * LSTR + kOff]);
            }
#pragma unroll
            for (int nt = 0; nt < 2; ++nt) {
                int cix = wc * 32 + nt * 16 + l15;
                bfrag[nt] = *(const v2f*)(&lb[cix * LSTR + kOff]);
            }
#pragma unroll
            for (int mt = 0; mt < 4; ++mt)
#pragma unroll
                for (int nt = 0; nt < 2; ++nt)
                    acc[mt][nt] = __builtin_amdgcn_wmma_f32_16x16x4_f32(
                        false, afrag[mt], false, bfrag[nt],
                        (short)0, acc[mt][nt], false, false);
        }
    }

    // Epilogue: d = sqrt(max(sq_i + sq_j - 2*gram, 0)); scatter lower triangle.
    // C/D layout: lane 0-15 N=l15 | VGPR v -> M=v (lanes<16) or M=v+8 (lanes>=16)
#pragma unroll
    for (int mt = 0; mt < 4; ++mt) {
#pragma unroll
        for (int nt = 0; nt < 2; ++nt) {
            const int j = colBase + wc * 32 + nt * 16 + l15;
            const float sqj = sq[j];
#pragma unroll
            for (int v = 0; v < 8; ++v) {
                const int i = rowBase + wr * 64 + mt * 16 + lhi * 8 + v;
                if (i > j) {
                    float d2 = sq[i] + sqj - 2.0f * acc[mt][nt][v];
                    d2 = d2 > 0.f ? d2 : 0.f;
                    long long idx = (((long long)i * (long long)(i - 1)) >> 1) + j;
                    out[idx] = __builtin_sqrtf(d2);
                }
            }
        }
    }
}

// ---------------- Launch ----------------
extern "C" void kernel_launch(void* const* d_in, const int* in_sizes, int n_in,
                              void* d_out, int out_size, void* d_ws, size_t ws_size,
                              hipStream_t stream) {
    const float* x = (const float*)d_in[0];
    float* out = (float*)d_out;
    float* sq  = (float*)d_ws;   // 4096 floats of scratch

    sqnorm_kernel<<<NPTS / 8, 256, 0, stream>>>(x, sq);

    dim3 grid(NPTS / BT, NPTS / BT);
    pairdist_kernel<<<grid, 256, 0, stream>>>(x, sq, out);
}